// MuleHunterGNN_50079318671880
// MI455X (gfx1250) — compile-verified
//
#include <hip/hip_runtime.h>
#include <hip/hip_bf16.h>
#include <math.h>

#define N_NODES 100000
#define N_EDGES 1600000
#define IN_F    20
#define H_F     64
#define HEADS   4
#define HO_F    32
#define HG_F    (HEADS * H_F)   // 256
#define EPS_BN  1e-5f

typedef float v2f __attribute__((ext_vector_type(2)));
typedef float v8f __attribute__((ext_vector_type(8)));

// ---------------------------------------------------------------- utilities
__global__ void k_fill(float* __restrict__ p, int n, float v) {
  int i = blockIdx.x * blockDim.x + threadIdx.x;
  if (i < n) p[i] = v;
}

__device__ inline void atomicMaxF(float* addr, float val) {
  unsigned int* ua = (unsigned int*)addr;
  unsigned int old = __float_as_uint(*addr);
  while (__uint_as_float(old) < val) {
    unsigned int assumed = old;
    old = atomicCAS(ua, assumed, __float_as_uint(val));
    if (old == assumed) break;
  }
}

// ---------------------------------------------------- skip branch (x@Wskip)
__global__ void k_identity(const float* __restrict__ x, const float* __restrict__ Wskip,
                           const float* __restrict__ bskip, float* __restrict__ ident) {
  int n = blockIdx.x * blockDim.x + threadIdx.x;
  if (n >= N_NODES) return;
  float acc[HO_F];
  #pragma unroll
  for (int c = 0; c < HO_F; ++c) acc[c] = bskip[c];
  for (int k = 0; k < IN_F; ++k) {
    float xv = x[n * IN_F + k];
    #pragma unroll
    for (int c = 0; c < HO_F; ++c) acc[c] += xv * Wskip[k * HO_F + c];
  }
  #pragma unroll
  for (int c = 0; c < HO_F; ++c) ident[n * HO_F + c] = acc[c];
}

// ------------------------------------------------------------- SAGE1 edges
__global__ void k_edge_aggr1(const int* __restrict__ src, const int* __restrict__ dst,
                             const float* __restrict__ x, float* __restrict__ aggr1,
                             float* __restrict__ deg) {
  int e = blockIdx.x * blockDim.x + threadIdx.x;
  if (e >= N_EDGES) return;
  int s = src[e], d = dst[e];
  atomicAdd(&deg[d], 1.0f);
  #pragma unroll
  for (int k = 0; k < IN_F; ++k)
    atomicAdd(&aggr1[d * IN_F + k], x[s * IN_F + k]);
}

// ------------------------------------------------------------- SAGE1 nodes
__global__ void k_sage1(const float* __restrict__ x, const float* __restrict__ aggr1,
                        const float* __restrict__ deg,
                        const float* __restrict__ W1l, const float* __restrict__ W1r,
                        const float* __restrict__ b1,
                        const float* __restrict__ g1, const float* __restrict__ be1,
                        const float* __restrict__ m1, const float* __restrict__ v1,
                        float* __restrict__ h1) {
  int i = blockIdx.x * blockDim.x + threadIdx.x;
  if (i >= N_NODES * H_F) return;
  int n = i / H_F, c = i % H_F;
  float invd = 1.0f / fmaxf(deg[n], 1.0f);
  float acc = b1[c];
  for (int k = 0; k < IN_F; ++k)
    acc += aggr1[n * IN_F + k] * invd * W1l[k * H_F + c] + x[n * IN_F + k] * W1r[k * H_F + c];
  float bn = (acc - m1[c]) * (g1[c] * rsqrtf(v1[c] + EPS_BN)) + be1[c];
  h1[i] = fmaxf(bn, 0.0f);
}

// --------------------------------------------- GAT projection via f32 WMMA
// hg[N,256] = h1[N,64] @ Wg[64,256].  One wave per 16x16 output tile.
// A(16x4) layout: lanes 0-15 hold row M=lane {K=k0,k0+1}; lanes 16-31 {K=k0+2,k0+3}.
// B(4x16): vgpr0 = rows K=k0 (lanes 0-15) / K=k0+2 (lanes 16-31); vgpr1 = K+1.
// C/D(16x16): vgpr r -> M = r (lanes 0-15), M = r+8 (lanes 16-31), N = lane&15.
__global__ void k_gemm_gat(const float* __restrict__ h1, const float* __restrict__ Wg,
                           float* __restrict__ hg) {
  int gid  = blockIdx.x * blockDim.x + threadIdx.x;
  int wave = gid >> 5;
  int lane = gid & 31;
  const int TN = HG_F / 16;           // 16 tiles along N
  int tm = wave / TN;
  int tn = wave % TN;
  if (tm >= N_NODES / 16) return;     // uniform per wave
  int m0 = tm * 16, n0 = tn * 16;
  int half = lane >> 4;               // 0 or 1
  int l    = lane & 15;
  v8f c = {0.f, 0.f, 0.f, 0.f, 0.f, 0.f, 0.f, 0.f};
  for (int k0 = 0; k0 < H_F; k0 += 4) {
    v2f a, b;
    int arow = m0 + l;
    a.x = h1[arow * H_F + k0 + 2 * half + 0];
    a.y = h1[arow * H_F + k0 + 2 * half + 1];
    b.x = Wg[(k0 + 2 * half + 0) * HG_F + n0 + l];
    b.y = Wg[(k0 + 2 * half + 1) * HG_F + n0 + l];
    c = __builtin_amdgcn_wmma_f32_16x16x4_f32(false, a, false, b, (short)0, c, false, false);
  }
  #pragma unroll
  for (int r = 0; r < 8; ++r)
    hg[(m0 + r + 8 * half) * HG_F + n0 + l] = c[r];
}

// ---------------------------------------------------- attention dot products
__global__ void k_att(const float* __restrict__ hg, const float* __restrict__ att_src,
                      const float* __restrict__ att_dst,
                      float* __restrict__ asrc, float* __restrict__ adst) {
  int i = blockIdx.x * blockDim.x + threadIdx.x;
  if (i >= N_NODES * HEADS) return;
  int n = i / HEADS, h = i % HEADS;
  float s = 0.f, d = 0.f;
  for (int k = 0; k < H_F; ++k) {
    float hv = hg[n * HG_F + h * H_F + k];
    s += hv * att_src[h * H_F + k];
    d += hv * att_dst[h * H_F + k];
  }
  asrc[i] = s;
  adst[i] = d;
}

// ------------------------------------------------------- segment max (edges)
__global__ void k_edge_max(const int* __restrict__ src, const int* __restrict__ dst,
                           const float* __restrict__ asrc, const float* __restrict__ adst,
                           float* __restrict__ mmax) {
  int e = blockIdx.x * blockDim.x + threadIdx.x;
  if (e >= N_EDGES) return;
  int s = src[e], d = dst[e];
  #pragma unroll
  for (int h = 0; h < HEADS; ++h) {
    float v = asrc[s * HEADS + h] + adst[d * HEADS + h];
    v = v > 0.f ? v : 0.2f * v;                     // leaky_relu 0.2
    atomicMaxF(&mmax[d * HEADS + h], v);
  }
}

// ----------------------------- softmax numerator/denominator (wave per edge)
__global__ void k_edge_acc(const int* __restrict__ src, const int* __restrict__ dst,
                           const float* __restrict__ asrc, const float* __restrict__ adst,
                           const float* __restrict__ mmax, const float* __restrict__ hg,
                           float* __restrict__ denom, float* __restrict__ gatacc) {
  int gid  = blockIdx.x * blockDim.x + threadIdx.x;
  int e    = gid >> 5;
  int lane = gid & 31;
  if (e >= N_EDGES) return;
  int s = src[e], d = dst[e];
  float ex[HEADS];
  #pragma unroll
  for (int h = 0; h < HEADS; ++h) {
    float v = asrc[s * HEADS + h] + adst[d * HEADS + h];
    v = v > 0.f ? v : 0.2f * v;
    ex[h] = expf(v - mmax[d * HEADS + h]);
  }
  if (lane == 0) {
    #pragma unroll
    for (int h = 0; h < HEADS; ++h) atomicAdd(&denom[d * HEADS + h], ex[h]);
  }
  #pragma unroll
  for (int h = 0; h < HEADS; ++h) {
    #pragma unroll
    for (int j = 0; j < 2; ++j) {
      int k = lane + 32 * j;
      atomicAdd(&gatacc[d * HG_F + h * H_F + k], ex[h] * hg[s * HG_F + h * H_F + k]);
    }
  }
}

// ----------------------------------------------------------- GAT node epilog
__global__ void k_gat_node(const float* __restrict__ gatacc, const float* __restrict__ denom,
                           const float* __restrict__ bg,
                           const float* __restrict__ g2, const float* __restrict__ be2,
                           const float* __restrict__ m2, const float* __restrict__ v2,
                           float* __restrict__ h2) {
  int i = blockIdx.x * blockDim.x + threadIdx.x;
  if (i >= N_NODES * H_F) return;
  int n = i / H_F, k = i % H_F;
  float acc = 0.f;
  #pragma unroll
  for (int h = 0; h < HEADS; ++h)
    acc += gatacc[n * HG_F + h * H_F + k] / (denom[n * HEADS + h] + 1e-16f);
  acc = acc * (1.0f / HEADS) + bg[k];
  float bn = (acc - m2[k]) * (g2[k] * rsqrtf(v2[k] + EPS_BN)) + be2[k];
  h2[i] = fmaxf(bn, 0.f);
}

// ------------------------------------------------- SAGE3 edges (wave per edge)
__global__ void k_edge_aggr3(const int* __restrict__ src, const int* __restrict__ dst,
                             const float* __restrict__ h2, float* __restrict__ aggr3) {
  int gid  = blockIdx.x * blockDim.x + threadIdx.x;
  int e    = gid >> 5;
  int lane = gid & 31;
  if (e >= N_EDGES) return;
  int s = src[e], d = dst[e];
  #pragma unroll
  for (int j = 0; j < 2; ++j) {
    int k = lane + 32 * j;
    atomicAdd(&aggr3[d * H_F + k], h2[s * H_F + k]);
  }
}

// ------------------------ SAGE3 node + skip + classifier + log_softmax fused
__global__ void k_final(const float* __restrict__ aggr3, const float* __restrict__ h2,
                        const float* __restrict__ deg, const float* __restrict__ ident,
                        const float* __restrict__ W3l, const float* __restrict__ W3r,
                        const float* __restrict__ b3,
                        const float* __restrict__ g3, const float* __restrict__ be3,
                        const float* __restrict__ m3, const float* __restrict__ v3,
                        const float* __restrict__ Wc1, const float* __restrict__ bc1,
                        const float* __restrict__ Wc2, const float* __restrict__ bc2,
                        float* __restrict__ out) {
  int n = blockIdx.x * blockDim.x + threadIdx.x;
  if (n >= N_NODES) return;
  float invd = 1.0f / fmaxf(deg[n], 1.0f);
  float h3[HO_F];
  #pragma unroll
  for (int c = 0; c < HO_F; ++c) h3[c] = b3[c];
  for (int k = 0; k < H_F; ++k) {
    float a = aggr3[n * H_F + k] * invd;
    float b = h2[n * H_F + k];
    #pragma unroll
    for (int c = 0; c < HO_F; ++c)
      h3[c] += a * W3l[k * HO_F + c] + b * W3r[k * HO_F + c];
  }
  #pragma unroll
  for (int c = 0; c < HO_F; ++c) {
    float bn = (h3[c] - m3[c]) * (g3[c] * rsqrtf(v3[c] + EPS_BN)) + be3[c];
    h3[c] = fmaxf(bn, 0.f) + ident[n * HO_F + c];
  }
  float l0 = bc2[0], l1 = bc2[1];
  #pragma unroll
  for (int j = 0; j < 32; ++j) {
    float acc = bc1[j];
    #pragma unroll
    for (int c = 0; c < HO_F; ++c) acc += h3[c] * Wc1[c * 32 + j];
    float z = fmaxf(acc, 0.f);
    l0 += z * Wc2[j * 2 + 0];
    l1 += z * Wc2[j * 2 + 1];
  }
  float mx  = fmaxf(l0, l1);
  float lse = mx + logf(expf(l0 - mx) + expf(l1 - mx));
  out[n * 2 + 0] = l0 - lse;
  out[n * 2 + 1] = l1 - lse;
}

// ---------------------------------------------------------------- launcher
extern "C" void kernel_launch(void* const* d_in, const int* in_sizes, int n_in,
                              void* d_out, int out_size, void* d_ws, size_t ws_size,
                              hipStream_t stream) {
  const float* x       = (const float*)d_in[0];
  const int*   ei      = (const int*)  d_in[1];
  const float* W1l     = (const float*)d_in[2];
  const float* W1r     = (const float*)d_in[3];
  const float* b1      = (const float*)d_in[4];
  const float* g1      = (const float*)d_in[5];
  const float* be1     = (const float*)d_in[6];
  const float* m1      = (const float*)d_in[7];
  const float* v1      = (const float*)d_in[8];
  const float* Wg      = (const float*)d_in[9];
  const float* att_src = (const float*)d_in[10];
  const float* att_dst = (const float*)d_in[11];
  const float* bg      = (const float*)d_in[12];
  const float* g2      = (const float*)d_in[13];
  const float* be2     = (const float*)d_in[14];
  const float* m2      = (const float*)d_in[15];
  const float* v2      = (const float*)d_in[16];
  const float* W3l     = (const float*)d_in[17];
  const float* W3r     = (const float*)d_in[18];
  const float* b3      = (const float*)d_in[19];
  const float* g3      = (const float*)d_in[20];
  const float* be3     = (const float*)d_in[21];
  const float* m3      = (const float*)d_in[22];
  const float* v3      = (const float*)d_in[23];
  const float* Wskip   = (const float*)d_in[24];
  const float* bskip   = (const float*)d_in[25];
  const float* Wc1     = (const float*)d_in[26];
  const float* bc1     = (const float*)d_in[27];
  const float* Wc2     = (const float*)d_in[28];
  const float* bc2     = (const float*)d_in[29];
  float* out = (float*)d_out;

  const int* src = ei;
  const int* dst = ei + N_EDGES;

  // workspace layout (floats) with lifetime reuse:
  //  regionA [N*256]: aggr1(@0,N*20) + h1(@N*20,N*64)  ->  gatacc  ->  aggr3
  //  hgbuf   [N*256]: hg  ->  h2
  float* ws      = (float*)d_ws;
  float* regionA = ws;
  float* aggr1   = regionA;
  float* h1      = regionA + (size_t)N_NODES * IN_F;
  float* gatacc  = regionA;
  float* aggr3   = regionA;
  float* hgbuf   = ws + (size_t)N_NODES * HG_F;
  float* hg      = hgbuf;
  float* h2      = hgbuf;
  float* ident   = ws + (size_t)N_NODES * (HG_F * 2);
  float* deg     = ident + (size_t)N_NODES * HO_F;
  float* asrc    = deg   + (size_t)N_NODES;
  float* adst    = asrc  + (size_t)N_NODES * HEADS;
  float* mmax    = adst  + (size_t)N_NODES * HEADS;
  float* denom   = mmax  + (size_t)N_NODES * HEADS;

  const int B = 256;
  auto blocks = [](int n) { return (n + 255) / 256; };

  // ---- SAGE1 ----
  k_fill<<<blocks(N_NODES), B, 0, stream>>>(deg, N_NODES, 0.f);
  k_fill<<<blocks(N_NODES * IN_F), B, 0, stream>>>(aggr1, N_NODES * IN_F, 0.f);
  k_identity<<<blocks(N_NODES), B, 0, stream>>>(x, Wskip, bskip, ident);
  k_edge_aggr1<<<blocks(N_EDGES), B, 0, stream>>>(src, dst, x, aggr1, deg);
  k_sage1<<<blocks(N_NODES * H_F), B, 0, stream>>>(x, aggr1, deg, W1l, W1r, b1,
                                                   g1, be1, m1, v1, h1);
  // ---- GAT projection (WMMA) ----
  int gemm_threads = (N_NODES / 16) * (HG_F / 16) * 32;
  k_gemm_gat<<<blocks(gemm_threads), B, 0, stream>>>(h1, Wg, hg);
  k_att<<<blocks(N_NODES * HEADS), B, 0, stream>>>(hg, att_src, att_dst, asrc, adst);
  // ---- GAT segment softmax ----
  k_fill<<<blocks(N_NODES * HEADS), B, 0, stream>>>(mmax, N_NODES * HEADS, -INFINITY);
  k_edge_max<<<blocks(N_EDGES), B, 0, stream>>>(src, dst, asrc, adst, mmax);
  k_fill<<<blocks(N_NODES * HEADS), B, 0, stream>>>(denom, N_NODES * HEADS, 0.f);
  k_fill<<<blocks(N_NODES * HG_F), B, 0, stream>>>(gatacc, N_NODES * HG_F, 0.f);
  k_edge_acc<<<blocks(N_EDGES * 32), B, 0, stream>>>(src, dst, asrc, adst, mmax, hg,
                                                     denom, gatacc);
  k_gat_node<<<blocks(N_NODES * H_F), B, 0, stream>>>(gatacc, denom, bg,
                                                      g2, be2, m2, v2, h2);
  // ---- SAGE3 + skip + classifier ----
  k_fill<<<blocks(N_NODES * H_F), B, 0, stream>>>(aggr3, N_NODES * H_F, 0.f);
  k_edge_aggr3<<<blocks(N_EDGES * 32), B, 0, stream>>>(src, dst, h2, aggr3);
  k_final<<<blocks(N_NODES), B, 0, stream>>>(aggr3, h2, deg, ident,
                                             W3l, W3r, b3, g3, be3, m3, v3,
                                             Wc1, bc1, Wc2, bc2, out);
}